// LineVectorizer_37297495998792
// MI455X (gfx1250) — compile-verified
//
#include <hip/hip_runtime.h>
#include <math.h>

typedef __attribute__((ext_vector_type(2))) float v2f;
typedef __attribute__((ext_vector_type(8))) float v8f;

#define HWPIX   65536
#define WDIM    256
#define C_IN    256
#define DIM_LOI 128
#define BATCH   4
#define NLINES  5000
#define NP1     8
#define DIM_FC  1024
#define NOUT    2500
#define M_ROWS  (BATCH * NLINES)   // 20000

// ---------------------------------------------------------------------------
// Stage 1: fc1 1x1 conv as WMMA f32 GEMM per batch.
// One wave -> one 16-pixel M-tile x ALL 8 N-tiles. Inner loop is explicitly
// double-buffered in registers: next k-step's A+B fragments are issued as a
// load batch before the current 8 wmmas consume theirs, so load latency
// overlaps matrix work instead of serializing on a single buffer pair.
// ---------------------------------------------------------------------------
__global__ void fc1_gemm_kernel(const float* __restrict__ feat,
                                const float* __restrict__ w,    // [128,256]
                                const float* __restrict__ bias, // [128]
                                float* __restrict__ xout) {     // [B,HW,128]
  int wave = blockIdx.x * (blockDim.x >> 5) + (threadIdx.x >> 5);
  int lane = threadIdx.x & 31;
  int b     = wave >> 12;        // 4096 M-tiles per batch
  int mtile = wave & 4095;

  int row = lane & 15;           // A row / B column (ISA frag layout)
  int khi = (lane >> 4) * 2;     // lanes 16-31 hold K+2, K+3
  int m0  = mtile * 16;

  const float* fbase = feat + (size_t)b * C_IN * HWPIX + m0 + row;
  const float* wbase = w + (size_t)row * C_IN + khi;

  v8f acc[8] = {};
  v2f a_cur, b_cur[8];
  a_cur.x = fbase[(size_t)khi * HWPIX];
  a_cur.y = fbase[(size_t)(khi + 1) * HWPIX];
#pragma unroll
  for (int t = 0; t < 8; ++t) b_cur[t] = *(const v2f*)(wbase + (size_t)t * 16 * C_IN);

  for (int k0 = 0; k0 < C_IN; k0 += 4) {
    int k1 = (k0 + 4 < C_IN) ? k0 + 4 : k0;   // last iter: harmless reload
    v2f a_nxt, b_nxt[8];
    a_nxt.x = fbase[(size_t)(k1 + khi) * HWPIX];
    a_nxt.y = fbase[(size_t)(k1 + khi + 1) * HWPIX];
#pragma unroll
    for (int t = 0; t < 8; ++t)
      b_nxt[t] = *(const v2f*)(wbase + (size_t)t * 16 * C_IN + k1);
#pragma unroll
    for (int t = 0; t < 8; ++t)
      acc[t] = __builtin_amdgcn_wmma_f32_16x16x4_f32(false, a_cur, false, b_cur[t],
                                                     (short)0, acc[t], false, false);
    a_cur = a_nxt;
#pragma unroll
    for (int t = 0; t < 8; ++t) b_cur[t] = b_nxt[t];
  }

  int ncol = lane & 15;
  int mhi  = (lane >> 4) * 8;    // C/D: lanes 16-31 hold M = r+8
  float* obase = xout + ((size_t)b * HWPIX + m0 + mhi) * DIM_LOI + ncol;
#pragma unroll
  for (int t = 0; t < 8; ++t) {
    float bs = bias[t * 16 + ncol];
#pragma unroll
    for (int r = 0; r < 8; ++r)
      obase[(size_t)r * DIM_LOI + t * 16] = acc[t][r] + bs;
  }
}

// ---------------------------------------------------------------------------
// Stage 2: bilinear sample 32 points / line from x[B,HW,128], maxpool(4),
// write flat[B*L, 1024] with layout f = c*8 + g (channel-major, pool-minor).
// One block (128 threads = one channel each) per line; every tap is a
// contiguous 512B read across the block.
// ---------------------------------------------------------------------------
__global__ void sample_pool_kernel(const float* __restrict__ x,
                                   const float* __restrict__ lines, // [B,L,2,2]
                                   float* __restrict__ flat) {
  int bl = blockIdx.x;            // 0..19999
  int b  = bl / NLINES;
  int c  = threadIdx.x;           // channel 0..127

  const float* lp = lines + (size_t)bl * 4;
  float p0x = lp[0], p0y = lp[1], p1x = lp[2], p1y = lp[3];
  const float* xb = x + (size_t)b * HWPIX * DIM_LOI;
  float* fout = flat + (size_t)bl * (DIM_LOI * NP1) + c * NP1;

  for (int g = 0; g < NP1; ++g) {
    float mx = -INFINITY;
#pragma unroll
    for (int s = 0; s < 4; ++s) {
      int p = g * 4 + s;
      float lam = (float)p * (1.0f / 31.0f);
      float px = p0x * lam + p1x * (1.0f - lam) - 0.5f;
      float py = p0y * lam + p1y * (1.0f - lam) - 0.5f;
      float px0 = fminf(fmaxf(floorf(px), 0.0f), 255.0f);
      float py0 = fminf(fmaxf(floorf(py), 0.0f), 255.0f);
      float px1 = fminf(px0 + 1.0f, 255.0f);
      float py1 = fminf(py0 + 1.0f, 255.0f);
      int ix0 = (int)px0, iy0 = (int)py0, ix1 = (int)px1, iy1 = (int)py1;
      float wx0 = px - px0, wx1 = px1 - px;
      float wy0 = py - py0, wy1 = py1 - py;
      float v = xb[((size_t)(ix0 * WDIM + iy0)) * DIM_LOI + c] * wx1 * wy1
              + xb[((size_t)(ix1 * WDIM + iy0)) * DIM_LOI + c] * wx0 * wy1
              + xb[((size_t)(ix0 * WDIM + iy1)) * DIM_LOI + c] * wx1 * wy0
              + xb[((size_t)(ix1 * WDIM + iy1)) * DIM_LOI + c] * wx0 * wy0;
      mx = fmaxf(mx, v);
    }
    fout[g] = mx;
  }
}

// ---------------------------------------------------------------------------
// Stage 3: MLP layer  out[M,N] = relu(A[M,K] * W[N,K]^T + bias)
// One wave -> one M-tile x 8 N-tiles; register double-buffered inner loop.
// ---------------------------------------------------------------------------
__global__ void mlp_gemm_kernel(const float* __restrict__ A,
                                const float* __restrict__ W,
                                const float* __restrict__ bias,
                                float* __restrict__ out,
                                int M, int N, int K, int do_relu) {
  const int NT = 8;
  int wave = blockIdx.x * (blockDim.x >> 5) + (threadIdx.x >> 5);
  int lane = threadIdx.x & 31;
  int ngroups = N / (16 * NT);
  int mtile = wave / ngroups;
  int ng    = wave - mtile * ngroups;
  if (mtile * 16 >= M) return;   // wave-uniform: EXEC stays all-ones for WMMA

  int row = lane & 15;
  int khi = (lane >> 4) * 2;
  int n0  = ng * 16 * NT;
  const float* abase = A + (size_t)(mtile * 16 + row) * K + khi;
  const float* wbase = W + (size_t)(n0 + row) * K + khi;

  v8f acc[NT] = {};
  v2f a_cur = *(const v2f*)(abase);
  v2f b_cur[NT];
#pragma unroll
  for (int t = 0; t < NT; ++t) b_cur[t] = *(const v2f*)(wbase + (size_t)t * 16 * K);

  for (int k0 = 0; k0 < K; k0 += 4) {
    int k1 = (k0 + 4 < K) ? k0 + 4 : k0;      // last iter: harmless reload
    v2f a_nxt = *(const v2f*)(abase + k1);
    v2f b_nxt[NT];
#pragma unroll
    for (int t = 0; t < NT; ++t)
      b_nxt[t] = *(const v2f*)(wbase + (size_t)t * 16 * K + k1);
#pragma unroll
    for (int t = 0; t < NT; ++t)
      acc[t] = __builtin_amdgcn_wmma_f32_16x16x4_f32(false, a_cur, false, b_cur[t],
                                                     (short)0, acc[t], false, false);
    a_cur = a_nxt;
#pragma unroll
    for (int t = 0; t < NT; ++t) b_cur[t] = b_nxt[t];
  }

  int ncol = lane & 15;
  int mhi  = (lane >> 4) * 8;
  float* obase = out + (size_t)(mtile * 16 + mhi) * N + n0 + ncol;
#pragma unroll
  for (int t = 0; t < NT; ++t) {
    float bs = bias[n0 + t * 16 + ncol];
#pragma unroll
    for (int r = 0; r < 8; ++r) {
      float v = acc[t][r] + bs;
      if (do_relu) v = fmaxf(v, 0.0f);
      obase[(size_t)r * N + t * 16] = v;
    }
  }
}

// ---------------------------------------------------------------------------
// Stage 4: head. One wave per line: 4 dots of length 1024, softmax, mask,
// sort key = mask ? argmax : -1.
// ---------------------------------------------------------------------------
__global__ void head_kernel(const float* __restrict__ H,   // [M,1024]
                            const float* __restrict__ w3,  // [4,1024]
                            const float* __restrict__ b3,  // [4]
                            float* __restrict__ scores,    // [M,4]
                            int* __restrict__ keys) {      // [M]
  int row  = blockIdx.x * (blockDim.x >> 5) + (threadIdx.x >> 5);
  int lane = threadIdx.x & 31;
  if (row >= M_ROWS) return;

  const float* h = H + (size_t)row * DIM_FC;
  float p0 = 0.f, p1 = 0.f, p2 = 0.f, p3 = 0.f;
  for (int k = lane; k < DIM_FC; k += 32) {
    float hv = h[k];
    p0 += hv * w3[k];
    p1 += hv * w3[DIM_FC + k];
    p2 += hv * w3[2 * DIM_FC + k];
    p3 += hv * w3[3 * DIM_FC + k];
  }
#pragma unroll
  for (int off = 16; off; off >>= 1) {
    p0 += __shfl_down(p0, off, 32);
    p1 += __shfl_down(p1, off, 32);
    p2 += __shfl_down(p2, off, 32);
    p3 += __shfl_down(p3, off, 32);
  }
  if (lane == 0) {
    float lg[4] = { p0 + b3[0], p1 + b3[1], p2 + b3[2], p3 + b3[3] };
    float mx = fmaxf(fmaxf(lg[0], lg[1]), fmaxf(lg[2], lg[3]));
    float e[4], sum = 0.f;
#pragma unroll
    for (int o = 0; o < 4; ++o) { e[o] = expf(lg[o] - mx); sum += e[o]; }
    float s[4];
#pragma unroll
    for (int o = 0; o < 4; ++o) s[o] = e[o] / sum;
    bool m = ((s[1] > 0.25f) || (s[2] > 0.25f) || (s[3] > 0.25f)) && (s[0] < 0.25f);
    int am = 0; float bv = s[0];
#pragma unroll
    for (int o = 1; o < 4; ++o) if (s[o] > bv) { bv = s[o]; am = o; }
    float* sp = scores + (size_t)row * 4;
    sp[0] = s[0]; sp[1] = s[1]; sp[2] = s[2]; sp[3] = s[3];
    keys[row] = m ? am : -1;
  }
}

// ---------------------------------------------------------------------------
// Stage 5: per-batch stable counting sort over keys {-1..3} (descending),
// cyclic pad to 2500. Matches jnp.argsort stability exactly.
// ---------------------------------------------------------------------------
__global__ void select_kernel(const float* __restrict__ lines,   // [B,L,2,2]
                              const float* __restrict__ scores,  // [B*L,4]
                              const int* __restrict__ keys,      // [B*L]
                              float* __restrict__ out_lines,     // [B,2500,2,2]
                              float* __restrict__ out_scores) {  // [B,2500,4]
  __shared__ int skeys[NLINES];
  __shared__ int order[NLINES];
  __shared__ int counts[4];
  __shared__ int offs[4];
  __shared__ int cnt_s;
  int b = blockIdx.x;
  int tid = threadIdx.x;

  if (tid < 4) counts[tid] = 0;
  __syncthreads();
  for (int i = tid; i < NLINES; i += blockDim.x) {
    int k = keys[b * NLINES + i];
    skeys[i] = k;
    if (k >= 0) atomicAdd(&counts[k], 1);
  }
  __syncthreads();
  if (tid == 0) {
    offs[3] = 0;
    offs[2] = counts[3];
    offs[1] = counts[3] + counts[2];
    offs[0] = counts[3] + counts[2] + counts[1];
    cnt_s   = counts[3] + counts[2] + counts[1] + counts[0];
  }
  __syncthreads();
  if (tid < 4) {                      // serial fill per bucket -> stable
    int key = 3 - tid;
    int pos = offs[key];
    for (int i = 0; i < NLINES; ++i)
      if (skeys[i] == key) order[pos++] = i;
  }
  __syncthreads();
  int cnt = cnt_s;
  for (int j = tid; j < NOUT; j += blockDim.x) {
    float l0 = 0, l1 = 0, l2 = 0, l3 = 0, s0 = 0, s1 = 0, s2 = 0, s3 = 0;
    if (cnt > 0) {
      int sel = order[j % cnt];
      const float* lp = lines + ((size_t)b * NLINES + sel) * 4;
      l0 = lp[0]; l1 = lp[1]; l2 = lp[2]; l3 = lp[3];
      const float* sp = scores + ((size_t)b * NLINES + sel) * 4;
      s0 = sp[0]; s1 = sp[1]; s2 = sp[2]; s3 = sp[3];
    }
    float* ol = out_lines + ((size_t)b * NOUT + j) * 4;
    ol[0] = l0; ol[1] = l1; ol[2] = l2; ol[3] = l3;
    float* os = out_scores + ((size_t)b * NOUT + j) * 4;
    os[0] = s0; os[1] = s1; os[2] = s2; os[3] = s3;
  }
}

// ---------------------------------------------------------------------------
extern "C" void kernel_launch(void* const* d_in, const int* in_sizes, int n_in,
                              void* d_out, int out_size, void* d_ws, size_t ws_size,
                              hipStream_t stream) {
  const float* feature = (const float*)d_in[0];
  const float* lines   = (const float*)d_in[1];
  const float* fc1_w   = (const float*)d_in[2];
  const float* fc1_b   = (const float*)d_in[3];
  const float* w1      = (const float*)d_in[4];
  const float* b1      = (const float*)d_in[5];
  const float* w2      = (const float*)d_in[6];
  const float* b2      = (const float*)d_in[7];
  const float* w3      = (const float*)d_in[8];
  const float* b3      = (const float*)d_in[9];

  float* out_lines  = (float*)d_out;                  // [4,2500,2,2]
  float* out_scores = out_lines + BATCH * NOUT * 4;   // [4,2500,4]

  // Workspace layout (x region reused for h1 after sampling): ~217 MB
  float* x      = (float*)d_ws;                           // B*HW*128
  float* flat   = x + (size_t)BATCH * HWPIX * DIM_LOI;    // 20000*1024
  float* h1     = x;                                      // alias: x dead after sampling
  float* scores = flat + (size_t)M_ROWS * DIM_FC;         // 20000*4
  int*   keys   = (int*)(scores + (size_t)M_ROWS * 4);    // 20000

  // Stage 1: fc1 GEMM. 4 * 4096 M-tile waves (full N per wave), 8 waves/block.
  fc1_gemm_kernel<<<(BATCH * 4096) / 8, 256, 0, stream>>>(feature, fc1_w, fc1_b, x);

  // Stage 2: bilinear sample + maxpool. One 128-thread block per line.
  sample_pool_kernel<<<M_ROWS, DIM_LOI, 0, stream>>>(x, lines, flat);

  // Stage 3: MLP. 1250 M-tiles * 8 N-groups = 10000 waves / layer.
  {
    int blocks = (1250 * 8) / 8;  // 1250
    mlp_gemm_kernel<<<blocks, 256, 0, stream>>>(flat, w1, b1, h1,
                                                M_ROWS, DIM_FC, DIM_FC, 1);
    mlp_gemm_kernel<<<blocks, 256, 0, stream>>>(h1, w2, b2, flat,
                                                M_ROWS, DIM_FC, DIM_FC, 1);
  }

  // Stage 4: head + softmax + keys. One wave per line, 8 waves/block.
  head_kernel<<<(M_ROWS + 7) / 8, 256, 0, stream>>>(flat, w3, b3, scores, keys);

  // Stage 5: per-batch stable select.
  select_kernel<<<BATCH, 256, 0, stream>>>(lines, scores, keys,
                                           out_lines, out_scores);
}